// FrechetSort_10746008175514
// MI455X (gfx1250) — compile-verified
//
#include <hip/hip_runtime.h>
#include <cstdint>
#include <cstddef>

#define NB      4096
#define NN      8192
#define THREADS 256
#define ITEMS   (NN / THREADS)          // 32 elements per thread
#define NEG_INF (-__builtin_inff())

typedef unsigned u32x4 __attribute__((ext_vector_type(4)));
typedef unsigned u32x8 __attribute__((ext_vector_type(8)));

// logaddexp(a,b); safe when exactly one arg is -inf.
__device__ __forceinline__ float lae(float a, float b) {
    float m = fmaxf(a, b);
    float d = fminf(a, b) - m;          // <= 0, or -inf
    return m + log1pf(__expf(d));
}

// Issue a TDM load of one contiguous row (NN x f32) into LDS.
// D# layout per CDNA5 ISA §8 (CDNA5 ISA).
__device__ __forceinline__ void tdm_load_row(const float* g_row, unsigned lds_byte_addr) {
    uint64_t ga = (uint64_t)(uintptr_t)g_row;
    u32x4 g0;
    g0[0] = 1u;                                   // count=1, is_restore=0, gather off
    g0[1] = lds_byte_addr;                        // lds_addr (bytes)
    g0[2] = (unsigned)(ga & 0xFFFFFFFFu);         // global_addr[31:0]
    g0[3] = (unsigned)((ga >> 32) & 0x01FFFFFFu)  // global_addr[56:32]
          | (2u << 30);                           // type=2 ("image")
    u32x8 g1;
    g1[0] = 0x00020000u;                          // workgroup_mask=0, data_size=2 (4B)
    g1[1] = (unsigned)(NN & 0xFFFFu) << 16;       // tensor_dim0[15:0] (8192)
    g1[2] = ((unsigned)(NN >> 16) & 0xFFFFu)      // tensor_dim0[31:16]
          | (1u << 16);                           // tensor_dim1[15:0] = 1
    g1[3] = (unsigned)(NN & 0xFFFFu) << 16;       // tensor_dim1 hi=0 | tile_dim0=8192
    g1[4] = 1u;                                   // tile_dim1=1, tile_dim2=0
    g1[5] = (unsigned)NN;                         // tensor_dim0_stride[31:0]
    g1[6] = 0u;                                   // stride hi, dim1_stride lo
    g1[7] = 0u;
    // 2-operand form: VADDR2/VADDR3 = NULL -> tensors up to 2D
    asm volatile("tensor_load_to_lds %0, %1" :: "s"(g0), "s"(g1) : "memory");
}

__global__ __launch_bounds__(THREADS)
void frechet_sort_kernel(const float* __restrict__ scores,
                         const float* __restrict__ gumbel_u,
                         float* __restrict__ out_action,
                         float* __restrict__ out_logp) {
    extern __shared__ __align__(16) char smem[];
    float* s_score = (float*)smem;                  // 32 KB: scores row (TDM)
    float* s_key   = (float*)(smem + NN * 4);       // 32 KB: u staging -> sort keys
    int*   s_idx   = (int*)  (smem + NN * 8);       // 32 KB: payload indices

    const unsigned tid   = threadIdx.x;
    const unsigned row   = blockIdx.x;
    const size_t   rbase = (size_t)row * NN;

    // ---- TDM: DMA the whole scores row into LDS (one descriptor, wave 0 only;
    //      TDM ignores EXEC, so gate by wave to avoid duplicate DMAs) ----
    if (tid < 32) {
        tdm_load_row(scores + rbase, (unsigned)(uintptr_t)s_score);
    }

    // ---- Async global->LDS: stream the gumbel_u row into the key region ----
    {
        unsigned long long sb = (unsigned long long)(uintptr_t)(gumbel_u + rbase);
        #pragma unroll
        for (int it = 0; it < ITEMS / 4; ++it) {    // 8 x b128 per thread
            unsigned e   = (unsigned)it * (THREADS * 4u) + tid * 4u;   // element idx
            unsigned lds = (unsigned)(uintptr_t)(s_key + e);           // LDS byte addr
            unsigned go  = e * 4u;                                     // global byte off
            asm volatile("global_load_async_to_lds_b128 %0, %1, %2 offset:0"
                         :: "v"(lds), "v"(go), "s"(sb) : "memory");
        }
    }
    asm volatile("s_wait_asynccnt 0" ::: "memory");
    __builtin_amdgcn_s_wait_tensorcnt(0);           // no-op for waves that issued none
    __syncthreads();

    // ---- Build (key, idx) in place: key = score - log(-log(u)) ----
    {
        const float4* sc4 = (const float4*)s_score;
        float4*       k4  = (float4*)s_key;         // holds u, overwritten with keys
        int4*         i4  = (int4*)s_idx;
        #pragma unroll
        for (int it = 0; it < ITEMS / 4; ++it) {
            unsigned q = (unsigned)it * THREADS + tid;
            float4 u = k4[q];
            float4 s = sc4[q];
            float4 k;
            k.x = s.x - logf(-logf(u.x));
            k.y = s.y - logf(-logf(u.y));
            k.z = s.z - logf(-logf(u.z));
            k.w = s.w - logf(-logf(u.w));
            k4[q] = k;
            int b0 = (int)(q * 4u);
            i4[q] = make_int4(b0, b0 + 1, b0 + 2, b0 + 3);
        }
    }

    // ---- Bitonic sort: strict total order (key desc, idx asc) == stable argsort(-key) ----
    for (unsigned k = 2; k <= NN; k <<= 1) {
        for (unsigned j = k >> 1; j > 0; j >>= 1) {
            __syncthreads();
            #pragma unroll 4
            for (unsigned p = tid; p < NN / 2; p += THREADS) {
                unsigned i = ((p & ~(j - 1u)) << 1) | (p & (j - 1u));
                unsigned m = i | j;
                float ka = s_key[i], kb = s_key[m];
                int   ia = s_idx[i], ib = s_idx[m];
                bool before = (ka > kb) || (ka == kb && ia < ib); // a precedes b (desc order)
                bool desc   = ((i & k) == 0u);
                if (before != desc) {
                    s_key[i] = kb; s_key[m] = ka;
                    s_idx[i] = ib; s_idx[m] = ia;
                }
            }
        }
    }
    __syncthreads();

    // ---- Emit action (coalesced; exact small ints in f32) ----
    #pragma unroll
    for (int it = 0; it < ITEMS; ++it) {
        unsigned e = (unsigned)it * THREADS + tid;
        out_action[rbase + e] = (float)s_idx[e];
    }

    // ---- log_proba: x = scores[action]; denom = suffix logcumsumexp; sum(x - denom) ----
    float x[ITEMS];
    {
        unsigned base = tid * ITEMS;                 // contiguous chunk per thread
        #pragma unroll
        for (int jj = 0; jj < ITEMS; ++jj)
            x[jj] = s_score[s_idx[base + jj]];       // SHAPE == 1.0
    }
    // chunk-local suffix logsumexp total
    float run = NEG_INF;
    #pragma unroll
    for (int jj = ITEMS - 1; jj >= 0; --jj) run = lae(run, x[jj]);

    __syncthreads();                                 // done reading s_idx
    float* s_red = (float*)s_idx;                    // reuse idx space
    s_red[tid] = run;
    __syncthreads();

    // inclusive suffix scan of chunk totals (log-step doubling)
    for (unsigned off = 1; off < THREADS; off <<= 1) {
        float v = s_red[tid];
        float w = (tid + off < THREADS) ? s_red[tid + off] : NEG_INF;
        __syncthreads();
        s_red[tid] = lae(v, w);
        __syncthreads();
    }
    float tail = (tid + 1 < THREADS) ? s_red[tid + 1] : NEG_INF;
    __syncthreads();

    // second reverse pass: denom_j = lae(local_suffix_j, tail)
    float part = 0.0f;
    run = NEG_INF;
    #pragma unroll
    for (int jj = ITEMS - 1; jj >= 0; --jj) {
        run = lae(run, x[jj]);
        part += x[jj] - lae(run, tail);
    }

    // block reduce partial sums
    s_red[tid] = part;
    __syncthreads();
    for (unsigned off = THREADS >> 1; off > 0; off >>= 1) {
        if (tid < off) s_red[tid] += s_red[tid + off];
        __syncthreads();
    }
    if (tid == 0) out_logp[row] = s_red[0];
}

extern "C" void kernel_launch(void* const* d_in, const int* in_sizes, int n_in,
                              void* d_out, int out_size, void* d_ws, size_t ws_size,
                              hipStream_t stream) {
    (void)in_sizes; (void)n_in; (void)out_size; (void)d_ws; (void)ws_size;
    const float* scores = (const float*)d_in[0];
    const float* gu     = (const float*)d_in[1];
    float* out_action = (float*)d_out;
    float* out_logp   = out_action + (size_t)NB * NN;

    const size_t shmem = (size_t)NN * 4 * 3;   // 96 KB per block (<= 320 KB/WGP)
    (void)hipFuncSetAttribute((const void*)frechet_sort_kernel,
                              hipFuncAttributeMaxDynamicSharedMemorySize,
                              (int)shmem);

    frechet_sort_kernel<<<NB, THREADS, shmem, stream>>>(scores, gu, out_action, out_logp);
}